// GradientTransform_9002251453028
// MI455X (gfx1250) — compile-verified
//
#include <hip/hip_runtime.h>
#include <math.h>

// Problem constants (from reference): B=8192, D=4096, RANK=16, HID=24
#define BROWS 8192
#define DDIM  4096
#define RANK_ 16
#define HID_  24
#define CHUNK 256           // D processed in 16 chunks of 256 columns
#define CH_STRIDE 260       // 260 mod 64 == 4 -> conflict-free WMMA A reads
#define TS_STRIDE 17        // 16x16 tile stride (17 mod 64 coprime enough)
#define HS_STRIDE 33
#define NROWBLK 64
#define ROWS_PER_BLK (BROWS / NROWBLK)

typedef __attribute__((ext_vector_type(2))) float v2f;
typedef __attribute__((ext_vector_type(8))) float v8f;

// ------------------------------------------------------------------
// Pass A: per-column partial sum / sum-of-squares over 128-row blocks
// ------------------------------------------------------------------
__global__ __launch_bounds__(256) void colstat_partial(const float* __restrict__ grad,
                                                       float* __restrict__ sums,
                                                       float* __restrict__ sqs) {
  const int c  = blockIdx.x * 256 + threadIdx.x;       // column (coalesced)
  const int r0 = blockIdx.y * ROWS_PER_BLK;
  float s = 0.f, q = 0.f;
  for (int r = r0; r < r0 + ROWS_PER_BLK; ++r) {
    const float g = grad[(size_t)r * DDIM + c];
    s += g;
    q = fmaf(g, g, q);
  }
  sums[(size_t)blockIdx.y * DDIM + c] = s;
  sqs [(size_t)blockIdx.y * DDIM + c] = q;
}

__global__ __launch_bounds__(256) void colstat_final(const float* __restrict__ sums,
                                                     const float* __restrict__ sqs,
                                                     float* __restrict__ meanp,
                                                     float* __restrict__ rinvp) {
  const int c = blockIdx.x * 256 + threadIdx.x;
  float s = 0.f, q = 0.f;
  for (int b = 0; b < NROWBLK; ++b) {
    s += sums[(size_t)b * DDIM + c];
    q += sqs [(size_t)b * DDIM + c];
  }
  const float mean = s / (float)BROWS;
  const float ssq  = q - (float)BROWS * mean * mean;   // sum((g-mean)^2)
  const float stdv = sqrtf(ssq / (float)(BROWS - 1));
  meanp[c] = mean;
  rinvp[c] = 1.f / (stdv + 1e-7f);
}

// ------------------------------------------------------------------
// WMMA helpers (V_WMMA_F32_16X16X4_F32, wave32)
// A 16x4 layout : lanes 0-15 hold M=0..15 {K0,K1}; lanes 16-31 {K2,K3}
// B 4x16 layout : lanes 0-15 hold N=0..15 {K0,K1}; lanes 16-31 {K2,K3}
// C/D 16x16     : VGPR j -> M = (lane<16 ? j : j+8), N = lane&15
// ------------------------------------------------------------------
__device__ __forceinline__ v8f wmma4(v2f a, v2f b, v8f c) {
  return __builtin_amdgcn_wmma_f32_16x16x4_f32(false, a, false, b, (short)0, c,
                                               false, false);
}

// acc += xsrc(16 x CHUNK, LDS) @ V[nrowoff+n][cb..cb+CHUNK)^T ; 8 k-steps/wave
__device__ __forceinline__ void rank_proj_acc(const float* __restrict__ xsrc,
                                              const float* __restrict__ V, // [nrows x DDIM]
                                              int cb, int nrowoff, int nrows,
                                              int wave, int lane, v8f& acc) {
  const int m    = lane & 15;
  const int n    = lane & 15;
  const int koff = (lane < 16) ? 0 : 2;
  const int vrow = nrowoff + n;
  const bool vok = (vrow < nrows);
  const float* vbase = V + (size_t)vrow * DDIM + cb;
#pragma unroll
  for (int ii = 0; ii < 8; ++ii) {
    const int k0 = (wave + 8 * ii) * 4;                 // 64 k-steps / chunk
    v2f a = *(const v2f*)&xsrc[m * CH_STRIDE + k0 + koff];
    v2f b = {};
    if (vok) b = *(const v2f*)&vbase[k0 + koff];
    acc = wmma4(a, b, acc);
  }
}

// dst(16 x CHUNK, LDS) = relu( ts(16x16) @ U[cb..][0..16)^T + bias ), 2 n-tiles/wave
__device__ __forceinline__ void rank_expand_chunk(const float* __restrict__ ts,
                                                  const float* __restrict__ U,    // [DDIM x 16]
                                                  const float* __restrict__ bias, // [DDIM]
                                                  float* __restrict__ dst,
                                                  int cb, int wave, int lane) {
  const int m    = lane & 15;
  const int n    = lane & 15;
  const int koff = (lane < 16) ? 0 : 2;
#pragma unroll
  for (int ti = 0; ti < 2; ++ti) {
    const int nt    = wave * 2 + ti;
    const int dcol  = nt * 16 + n;
    const int dglob = cb + dcol;
    v8f acc = {};
#pragma unroll
    for (int kb = 0; kb < 4; ++kb) {                    // K = 16 = 4 WMMA steps
      v2f a;
      a.x = ts[m * TS_STRIDE + kb * 4 + koff];
      a.y = ts[m * TS_STRIDE + kb * 4 + koff + 1];
      v2f b = *(const v2f*)&U[(size_t)dglob * RANK_ + kb * 4 + koff];
      acc = wmma4(a, b, acc);
    }
    const float bv = bias[dglob];
#pragma unroll
    for (int j = 0; j < 8; ++j) {
      const int M = (lane < 16) ? j : (j + 8);
      dst[M * CH_STRIDE + dcol] = fmaxf(acc[j] + bv, 0.f);
    }
  }
}

// reduce 8 per-wave 16x16 WMMA accumulators -> dst (stride dstride)
__device__ __forceinline__ void reduce_tile(const v8f& acc, float* __restrict__ wacc,
                                            float* __restrict__ dst, int dstride,
                                            int wave, int lane, int t) {
#pragma unroll
  for (int j = 0; j < 8; ++j) {
    const int M = (lane < 16) ? j : (j + 8);
    const int N = lane & 15;
    wacc[wave * 256 + M * 16 + N] = acc[j];
  }
  __syncthreads();
  float s = 0.f;
#pragma unroll
  for (int w = 0; w < 8; ++w) s += wacc[w * 256 + t];
  dst[(t >> 4) * dstride + (t & 15)] = s;
  __syncthreads();
}

__device__ __forceinline__ float wave_sum(float v) {
#pragma unroll
  for (int off = 16; off > 0; off >>= 1) v += __shfl_down(v, off, 32);
  return v;
}

// ------------------------------------------------------------------
// Fused main kernel: one 16-row tile per 256-thread (8-wave) block
// ------------------------------------------------------------------
__global__ __launch_bounds__(256) void fused_gt(
    const float* __restrict__ grad, const float* __restrict__ embed,
    const float* __restrict__ u0, const float* __restrict__ v0, const float* __restrict__ b0,
    const float* __restrict__ u1, const float* __restrict__ v1, const float* __restrict__ b1,
    const float* __restrict__ w1, const float* __restrict__ bw1,
    const float* __restrict__ w2, const float* __restrict__ bw2,
    const float* __restrict__ meanp, const float* __restrict__ rinvp,
    float* __restrict__ out, float* __restrict__ coeff_out) {
  __shared__ float xs[16 * CH_STRIDE];   // x0 / embed staging
  __shared__ float ys[16 * CH_STRIDE];   // x1 / x2 staging
  __shared__ float wacc[8 * 256];        // cross-wave WMMA reduce
  __shared__ float t0s[16 * TS_STRIDE];  // t0 = x0 @ v0^T
  __shared__ float t1s[16 * TS_STRIDE];  // t1 = x1 @ v1^T
  __shared__ float hs[16 * HS_STRIDE];   // h  = embed @ w1^T (16x32, 24 used)
  __shared__ float x0n_s[16], xn_s[16], c_s[16], scale_s[16];

  const int t = threadIdx.x;
  const int lane = t & 31;
  const int wave = t >> 5;
  const int rowbase = blockIdx.x * 16;

  if (t < 16) { x0n_s[t] = 0.f; xn_s[t] = 0.f; }
  __syncthreads();

  // ---- pass 1: x0 = (g - mean)*rinv ; t0 += x0 @ v0^T ; ||x0||^2 ----
  v8f t0acc = {};
  float x0np[16];
#pragma unroll
  for (int i = 0; i < 16; ++i) x0np[i] = 0.f;
  for (int cb = 0; cb < DDIM; cb += CHUNK) {
    const int col = cb + t;
    const float mc = meanp[col], rc = rinvp[col];
#pragma unroll
    for (int i = 0; i < 16; ++i) {
      const float g  = grad[(size_t)(rowbase + i) * DDIM + col];
      const float x0 = (g - mc) * rc;
      xs[i * CH_STRIDE + t] = x0;
      x0np[i] = fmaf(x0, x0, x0np[i]);
    }
    if (cb + CHUNK < DDIM)
      __builtin_prefetch(&grad[(size_t)rowbase * DDIM + cb + CHUNK + t], 0, 0);
    __syncthreads();
    rank_proj_acc(xs, v0, cb, 0, RANK_, wave, lane, t0acc);
    __syncthreads();
  }
  reduce_tile(t0acc, wacc, t0s, TS_STRIDE, wave, lane, t);
#pragma unroll
  for (int i = 0; i < 16; ++i) {
    const float v = wave_sum(x0np[i]);
    if (lane == 0) atomicAdd(&x0n_s[i], v);
  }

  // ---- pass 2: x1 = relu(t0 @ u0^T + b0) ; t1 += x1 @ v1^T ----
  v8f t1acc = {};
  for (int cb = 0; cb < DDIM; cb += CHUNK) {
    rank_expand_chunk(t0s, u0, b0, ys, cb, wave, lane);
    __syncthreads();
    rank_proj_acc(ys, v1, cb, 0, RANK_, wave, lane, t1acc);
    __syncthreads();
  }
  reduce_tile(t1acc, wacc, t1s, TS_STRIDE, wave, lane, t);

  // ---- pass 3: x2 = relu(t1 @ u1^T + b1) ; ||x2||^2 ----
  float xnp[16];
#pragma unroll
  for (int i = 0; i < 16; ++i) xnp[i] = 0.f;
  for (int cb = 0; cb < DDIM; cb += CHUNK) {
    rank_expand_chunk(t1s, u1, b1, ys, cb, wave, lane);
    __syncthreads();
#pragma unroll
    for (int i = 0; i < 16; ++i) {
      const float v = ys[i * CH_STRIDE + t];
      xnp[i] = fmaf(v, v, xnp[i]);
    }
    __syncthreads();
  }
#pragma unroll
  for (int i = 0; i < 16; ++i) {
    const float v = wave_sum(xnp[i]);
    if (lane == 0) atomicAdd(&xn_s[i], v);
  }

  // ---- pass 4: h = embed @ w1^T (two 16-wide n-tiles, rows>=24 zeroed) ----
  v8f h0 = {}, h1 = {};
  for (int cb = 0; cb < DDIM; cb += CHUNK) {
#pragma unroll
    for (int i = 0; i < 16; ++i)
      xs[i * CH_STRIDE + t] = embed[(size_t)(rowbase + i) * DDIM + cb + t];
    if (cb + CHUNK < DDIM)
      __builtin_prefetch(&embed[(size_t)rowbase * DDIM + cb + CHUNK + t], 0, 0);
    __syncthreads();
    rank_proj_acc(xs, w1, cb, 0,  HID_, wave, lane, h0);
    rank_proj_acc(xs, w1, cb, 16, HID_, wave, lane, h1);
    __syncthreads();
  }
  reduce_tile(h0, wacc, hs,      HS_STRIDE, wave, lane, t);
  reduce_tile(h1, wacc, hs + 16, HS_STRIDE, wave, lane, t);

  // ---- coeff (sigmoid MLP head) + per-row rescale factors ----
  if (t < 16) {
    float s = bw2[0];
#pragma unroll
    for (int j = 0; j < HID_; ++j)
      s = fmaf(fmaxf(hs[t * HS_STRIDE + j] + bw1[j], 0.f), w2[j], s);
    const float c = 1.f / (1.f + __expf(-s));
    c_s[t] = c;
    coeff_out[rowbase + t] = c;
    const float xn  = sqrtf(xn_s[t])  + 1e-8f;
    const float x0n = sqrtf(x0n_s[t]) + 1e-8f;
    scale_s[t] = x0n / xn;
  }
  __syncthreads();

  // ---- pass 5: out = (1-c)*scale*x2 + c*x0  (grad tile re-read hits L2) ----
  for (int cb = 0; cb < DDIM; cb += CHUNK) {
    const int col = cb + t;
    const float mc = meanp[col], rc = rinvp[col], bc = b1[col];
    float u1c[RANK_];
#pragma unroll
    for (int k = 0; k < RANK_; ++k) u1c[k] = u1[(size_t)col * RANK_ + k];
#pragma unroll
    for (int i = 0; i < 16; ++i) {
      const float g  = grad[(size_t)(rowbase + i) * DDIM + col];
      const float x0 = (g - mc) * rc;
      float x2 = bc;
#pragma unroll
      for (int k = 0; k < RANK_; ++k)
        x2 = fmaf(t1s[i * TS_STRIDE + k], u1c[k], x2);
      x2 = fmaxf(x2, 0.f);
      const float c = c_s[i];
      out[(size_t)(rowbase + i) * DDIM + col] =
          (1.f - c) * (x2 * scale_s[i]) + c * x0;
    }
  }
}

// ------------------------------------------------------------------
extern "C" void kernel_launch(void* const* d_in, const int* in_sizes, int n_in,
                              void* d_out, int out_size, void* d_ws, size_t ws_size,
                              hipStream_t stream) {
  const float* grad  = (const float*)d_in[0];
  const float* embed = (const float*)d_in[1];
  const float* u0    = (const float*)d_in[2];
  const float* v0    = (const float*)d_in[3];
  const float* b0    = (const float*)d_in[4];
  const float* u1    = (const float*)d_in[5];
  const float* v1    = (const float*)d_in[6];
  const float* b1    = (const float*)d_in[7];
  const float* w1    = (const float*)d_in[8];
  const float* bw1   = (const float*)d_in[9];
  const float* w2    = (const float*)d_in[10];
  const float* bw2   = (const float*)d_in[11];

  float* ws    = (float*)d_ws;
  float* sums  = ws;                                  // 64*4096
  float* sqs   = ws + (size_t)NROWBLK * DDIM;         // 64*4096
  float* meanp = ws + (size_t)2 * NROWBLK * DDIM;     // 4096
  float* rinvp = meanp + DDIM;                        // 4096  (~2.03 MB total)

  float* out   = (float*)d_out;
  float* coeff = out + (size_t)BROWS * DDIM;

  colstat_partial<<<dim3(DDIM / 256, NROWBLK), 256, 0, stream>>>(grad, sums, sqs);
  colstat_final<<<DDIM / 256, 256, 0, stream>>>(sums, sqs, meanp, rinvp);
  fused_gt<<<BROWS / 16, 256, 0, stream>>>(grad, embed, u0, v0, b0, u1, v1, b1,
                                           w1, bw1, w2, bw2, meanp, rinvp, out, coeff);
}